// GATBackbone_89550068122077
// MI455X (gfx1250) — compile-verified
//
#include <hip/hip_runtime.h>
#include <hip/hip_bf16.h>

typedef float v2f __attribute__((ext_vector_type(2)));
typedef float v8f __attribute__((ext_vector_type(8)));

#define CCH 128   // per-head channels, fixed by the model
#define KCH 16    // K-chunk staged in LDS per block
#define LPAD 18   // LDS column stride (floats): even (b64-aligned), conflict-spreading

// ---------------- WMMA f32 GEMM: Y[nrows x dout] = X[nrows x din] @ W[din x dout]
// Block = 8 waves x 16 rows = 128 rows, 64 cols. W k-chunk staged in LDS, shared by
// all 8 waves; each wave holds one A fragment per k-step and issues 4 WMMAs.
// A frag (16x4 f32, 2 VGPR/lane): v0 = A[m][k+kk], v1 = A[m][k+kk+1], kk = (lane>=16)*2
// C/D frag (16x16 f32, 8 VGPR/lane): vr = D[rowBase + r + (lane>=16)*8][colBase + (lane&15)]
__global__ void gemm_wmma_f32(const float* __restrict__ X, const float* __restrict__ W,
                              float* __restrict__ Y, int nrows, int din, int dout) {
  __shared__ float ws[64 * LPAD];
  const int tid  = threadIdx.x;
  const int wave = tid >> 5;
  const int lane = tid & 31;
  int rowBase = (blockIdx.x * 8 + wave) * 16;
  if (rowBase > nrows - 16) rowBase = nrows - 16;  // tail waves duplicate last tile (identical values)
  const int colBase = blockIdx.y * 64;

  v8f acc0 = {}, acc1 = {}, acc2 = {}, acc3 = {};
  const int m  = rowBase + (lane & 15);
  const int kk = (lane >> 4) * 2;
  const int c  = (lane & 15);
  const float* __restrict__ xrow = X + (size_t)m * din;

  const int scol = tid & 63;   // staging column 0..63
  const int sk0  = tid >> 6;   // staging k row 0..3

  for (int k0 = 0; k0 < din; k0 += KCH) {
    // stage W[k0..k0+15][colBase..colBase+63] into LDS as ws[col*LPAD + k]
#pragma unroll
    for (int rr = 0; rr < KCH; rr += 4) {
      const int kr = sk0 + rr;
      ws[scol * LPAD + kr] = W[(size_t)(k0 + kr) * dout + colBase + scol];
    }
    __syncthreads();
#pragma unroll
    for (int kc = 0; kc < KCH; kc += 4) {
      v2f a;
      a.x = xrow[k0 + kc + kk];
      a.y = xrow[k0 + kc + kk + 1];
      const float* bp = &ws[c * LPAD + kc + kk];     // 8B-aligned: c*18, kc, kk all even
      v2f b0 = *(const v2f*)(bp);
      v2f b1 = *(const v2f*)(bp + 16 * LPAD);
      v2f b2 = *(const v2f*)(bp + 32 * LPAD);
      v2f b3 = *(const v2f*)(bp + 48 * LPAD);
      acc0 = __builtin_amdgcn_wmma_f32_16x16x4_f32(false, a, false, b0, (short)0, acc0, false, false);
      acc1 = __builtin_amdgcn_wmma_f32_16x16x4_f32(false, a, false, b1, (short)0, acc1, false, false);
      acc2 = __builtin_amdgcn_wmma_f32_16x16x4_f32(false, a, false, b2, (short)0, acc2, false, false);
      acc3 = __builtin_amdgcn_wmma_f32_16x16x4_f32(false, a, false, b3, (short)0, acc3, false, false);
    }
    __syncthreads();
  }

  const int rofs = (lane >> 4) * 8;
  float* __restrict__ yb = Y + (size_t)(rowBase + rofs) * dout + colBase + c;
#pragma unroll
  for (int r = 0; r < 8; ++r) {
    float* yr = yb + (size_t)r * dout;
    yr[0]  = acc0[r];
    yr[16] = acc1[r];
    yr[32] = acc2[r];
    yr[48] = acc3[r];
  }
}

// ---------------- attention projections: a_src[n,h] = <xl[n,h,:], att_src[h,:]>
__global__ void att_dot(const float* __restrict__ xl,
                        const float* __restrict__ att_s, const float* __restrict__ att_d,
                        float* __restrict__ asn, float* __restrict__ adn, int n, int H) {
  int t = blockIdx.x * blockDim.x + threadIdx.x;
  if (t >= n * H) return;
  int node = t / H, h = t - node * H;
  const float* __restrict__ xp = xl + ((size_t)node * H + h) * CCH;
  const float* __restrict__ sp = att_s + h * CCH;
  const float* __restrict__ dp = att_d + h * CCH;
  float ss = 0.f, sd = 0.f;
#pragma unroll 4
  for (int cc = 0; cc < CCH; ++cc) { float v = xp[cc]; ss += v * sp[cc]; sd += v * dp[cc]; }
  asn[t] = ss;
  adn[t] = sd;
}

// monotonic uint encoding of float for integer atomicMax
__device__ __forceinline__ unsigned encF(float v) {
  unsigned u = __float_as_uint(v);
  return (u & 0x80000000u) ? ~u : (u | 0x80000000u);
}
__device__ __forceinline__ float decF(unsigned e) {
  unsigned u = (e & 0x80000000u) ? (e & 0x7FFFFFFFu) : ~e;
  return __uint_as_float(u);
}

__global__ void init_mdenom(unsigned* __restrict__ menc, float* __restrict__ denom, int n) {
  int t = blockIdx.x * blockDim.x + threadIdx.x;
  if (t < n) { menc[t] = 0x007FFFFFu; /* enc(-inf) */ denom[t] = 0.f; }
}

// per (edge, head): alpha = leaky_relu(a_src[s]+a_dst[d]); segment-max into menc[dst]
__global__ void edge_alpha_max(const int* __restrict__ ei, int E, int Etot, int H,
                               const float* __restrict__ asn, const float* __restrict__ adn,
                               float* __restrict__ alpha, unsigned* __restrict__ menc) {
  int t = blockIdx.x * blockDim.x + threadIdx.x;
  if (t >= Etot * H) return;
  int e = t / H, h = t - e * H;
  int s, d;
  if (e < E) { s = ei[e]; d = ei[E + e]; } else { s = e - E; d = s; }
  float al = asn[s * H + h] + adn[d * H + h];
  al = al > 0.f ? al : 0.2f * al;
  alpha[t] = al;
  atomicMax(&menc[d * H + h], encF(al));
}

// per (edge, head): ea = exp(alpha - m[dst]); denom[dst] += ea (in place over alpha buf)
__global__ void edge_exp_sum(const int* __restrict__ ei, int E, int Etot, int H,
                             const unsigned* __restrict__ menc,
                             float* __restrict__ alpha, float* __restrict__ denom) {
  int t = blockIdx.x * blockDim.x + threadIdx.x;
  if (t >= Etot * H) return;
  int e = t / H, h = t - e * H;
  int d = (e < E) ? ei[E + e] : (e - E);
  float m  = decF(menc[d * H + h]);
  float ea = __expf(alpha[t] - m);
  alpha[t] = ea;
  atomicAdd(&denom[d * H + h], ea);
}

// one wave per edge: agg[dst] += xl[src] * (ea/denom[dst]) , float4 gather + f32 atomics
__global__ void edge_aggregate(const int* __restrict__ ei, int E, int Etot, int H, int dout,
                               const float* __restrict__ ea, const float* __restrict__ denom,
                               const float* __restrict__ xl, float* __restrict__ agg) {
  const int wave = threadIdx.x >> 5;
  const int lane = threadIdx.x & 31;
  int e = blockIdx.x * (blockDim.x >> 5) + wave;
  if (e >= Etot) return;
  int s, d;
  if (e < E) { s = ei[e]; d = ei[E + e]; } else { s = e - E; d = s; }
  float wgt[4];
#pragma unroll
  for (int h = 0; h < 4; ++h)
    if (h < H) wgt[h] = ea[e * H + h] / (denom[d * H + h] + 1e-16f);
  const float4* __restrict__ xs = (const float4*)(xl + (size_t)s * dout);
  float* __restrict__ ad = agg + (size_t)d * dout;
  const int nvec = dout >> 2;
  for (int i = lane; i < nvec; i += 32) {      // head = (i*4)/128 = i>>5
    float4 v = xs[i];
    float w = wgt[i >> 5];
    atomicAdd(ad + i * 4 + 0, v.x * w);
    atomicAdd(ad + i * 4 + 1, v.y * w);
    atomicAdd(ad + i * 4 + 2, v.z * w);
    atomicAdd(ad + i * 4 + 3, v.w * w);
  }
}

// x = elu( ((agg + bias) - mean) * rsqrt(var+eps) * gamma + beta )
__global__ void bias_bn_elu(const float* __restrict__ agg, const float* __restrict__ bias,
                            const float* __restrict__ gamma, const float* __restrict__ beta,
                            const float* __restrict__ mean, const float* __restrict__ var,
                            float* __restrict__ out, int n, int dout) {
  int t = blockIdx.x * blockDim.x + threadIdx.x;
  if (t >= n * dout) return;
  int cc = t % dout;
  float v = agg[t] + bias[cc];
  v = (v - mean[cc]) * rsqrtf(var[cc] + 1e-5f) * gamma[cc] + beta[cc];
  out[t] = v > 0.f ? v : (__expf(v) - 1.f);
}

__global__ void pool_sum(const float* __restrict__ x, const int* __restrict__ bidx,
                         float* __restrict__ pooled, float* __restrict__ cnt, int n) {
  int t = blockIdx.x * blockDim.x + threadIdx.x;
  if (t >= n * CCH) return;
  int node = t / CCH, cc = t - node * CCH;
  int b = bidx[node];
  atomicAdd(&pooled[b * CCH + cc], x[t]);
  if (cc == 0) atomicAdd(&cnt[b], 1.f);
}

__global__ void pool_div(const float* __restrict__ pooled, const float* __restrict__ cnt,
                         float* __restrict__ out, int total) {
  int t = blockIdx.x * blockDim.x + threadIdx.x;
  if (t >= total) return;
  out[t] = pooled[t] / fmaxf(cnt[t / CCH], 1.f);
}

extern "C" void kernel_launch(void* const* d_in, const int* in_sizes, int n_in,
                              void* d_out, int out_size, void* d_ws, size_t ws_size,
                              hipStream_t stream) {
  const float* x_in = (const float*)d_in[0];
  const int* ei     = (const int*)d_in[1];
  const int* bidx   = (const int*)d_in[2];

  const int N    = in_sizes[2];
  const int E    = in_sizes[1] / 2;
  const int Etot = E + N;
  const int IN_  = in_sizes[0] / N;
  const int G    = out_size / CCH;
  const int MAXH = 4;
  const size_t MAXD = 512;

  // workspace partition (floats)
  float*    B0     = (float*)d_ws;                 // xl  (GEMM out)
  float*    B1     = B0 + (size_t)N * MAXD;        // agg (scatter target)
  float*    B2     = B1 + (size_t)N * MAXD;        // x for next layer
  float*    asn    = B2 + (size_t)N * MAXD;        // [N,H]
  float*    adn    = asn + (size_t)N * MAXH;
  unsigned* menc   = (unsigned*)(adn + (size_t)N * MAXH);
  float*    denom  = (float*)menc + (size_t)N * MAXH;
  float*    eaBuf  = denom + (size_t)N * MAXH;     // [Etot,H]
  float*    pooled = eaBuf + (size_t)Etot * MAXH;  // [G,128]
  float*    cnt    = pooled + (size_t)G * CCH;     // [G]

  const int TB = 256;
  int din = IN_;
  const float* xin = x_in;

  for (int l = 0; l < 3; ++l) {
    const int base = 3 + 8 * l;
    const float* W     = (const float*)d_in[base + 0];
    const float* att_s = (const float*)d_in[base + 1];
    const float* att_d = (const float*)d_in[base + 2];
    const float* bias  = (const float*)d_in[base + 3];
    const float* gamma = (const float*)d_in[base + 4];
    const float* beta  = (const float*)d_in[base + 5];
    const float* mean  = (const float*)d_in[base + 6];
    const float* var   = (const float*)d_in[base + 7];
    const int dout = in_sizes[base + 0] / din;
    const int H    = in_sizes[base + 1] / CCH;

    // 1) xl = x @ W  (WMMA f32, LDS-staged W)
    {
      int rowTiles = (N + 15) / 16;
      dim3 grid((rowTiles + 7) / 8, dout / 64);
      gemm_wmma_f32<<<grid, TB, 0, stream>>>(xin, W, B0, N, din, dout);
    }
    // 2) attention projections
    att_dot<<<(N * H + TB - 1) / TB, TB, 0, stream>>>(B0, att_s, att_d, asn, adn, N, H);
    // 3) segment softmax over dst
    init_mdenom<<<(N * H + TB - 1) / TB, TB, 0, stream>>>(menc, denom, N * H);
    edge_alpha_max<<<(Etot * H + TB - 1) / TB, TB, 0, stream>>>(ei, E, Etot, H, asn, adn, eaBuf, menc);
    edge_exp_sum<<<(Etot * H + TB - 1) / TB, TB, 0, stream>>>(ei, E, Etot, H, menc, eaBuf, denom);
    // 4) weighted scatter aggregation
    hipMemsetAsync(B1, 0, (size_t)N * dout * sizeof(float), stream);
    edge_aggregate<<<(Etot + 7) / 8, TB, 0, stream>>>(ei, E, Etot, H, dout, eaBuf, denom, B0, B1);
    // 5) bias + BN(eval) + ELU
    bias_bn_elu<<<(N * dout + TB - 1) / TB, TB, 0, stream>>>(B1, bias, gamma, beta, mean, var, B2, N, dout);

    xin = B2;
    din = dout;
  }

  // global mean pool
  hipMemsetAsync(pooled, 0, (size_t)(G * CCH + G) * sizeof(float), stream);
  pool_sum<<<(N * CCH + TB - 1) / TB, TB, 0, stream>>>(B2, bidx, pooled, cnt, N);
  pool_div<<<(out_size + TB - 1) / TB, TB, 0, stream>>>(pooled, cnt, (float*)d_out, out_size);
}